// WaveFunctionDensity_25786983645324
// MI455X (gfx1250) — compile-verified
//
#include <hip/hip_runtime.h>

// WaveFunctionDensity for MI455X (gfx1250, wave32).
// rho[n,m] = basis[n,m,:] @ dm[n] @ basis[n,m,:]  via V_WMMA_F32_16X16X4_F32.

typedef __attribute__((ext_vector_type(2))) float v2f;
typedef __attribute__((ext_vector_type(8))) float v8f;

#define NMOL 16
#define MS   4096
#define NA   32
#define NP   512
#define MT   2              // 16-row sample tiles per wave (32 samples/wave)
#define TILEGRPS (MS / (16 * MT))   // 128 tile groups per molecule

// Packed WFN symmetry table: px | py<<4 | pz<<8
__device__ __constant__ unsigned int SYMPK[20] = {
    0x000,
    0x001, 0x010, 0x100,
    0x002, 0x020, 0x200, 0x011, 0x101, 0x110,
    0x003, 0x030, 0x300, 0x012, 0x102, 0x120,
    0x021, 0x201, 0x210, 0x111,
};

__device__ __forceinline__ float powi3(float x, unsigned e) {
    // x^e for e in {0,1,2,3}; pow(0,0)==1 matches XLA integral-exponent pow.
    float r = 1.0f;
    if (e >= 1u) r = x;
    if (e >= 2u) r *= x;
    if (e >= 3u) r *= x;
    return r;
}

__global__ __launch_bounds__(32)
void wfn_density_wmma_kernel(const float* __restrict__ dv,
                             const int*   __restrict__ centers,
                             const float* __restrict__ exps,
                             const int*   __restrict__ syms,
                             const float* __restrict__ dm,
                             float*       __restrict__ out) {
    extern __shared__ float smem[];
    float* bas = smem;                       // [MT][NP][16] fp32  (64 KB)
    float* dvs = smem + MT * NP * 16;        // [32][NA][4] x,y,z,r2 (16 KB)

    const int lane  = threadIdx.x;           // 0..31, one wave32
    const int mol   = blockIdx.x / TILEGRPS;
    const int tg    = blockIdx.x % TILEGRPS;
    const int sbase = tg * (16 * MT);
    const int samp  = sbase + lane;          // this lane's sample point

    // ---- Phase 0: cache this sample's 32 atom displacement vectors + r^2 ----
    const float* dvm = dv + (size_t)(mol * MS + samp) * (NA * 3);
    #pragma unroll 4
    for (int a = 0; a < NA; ++a) {
        float x = dvm[a * 3 + 0];
        float y = dvm[a * 3 + 1];
        float z = dvm[a * 3 + 2];
        float r2 = x * x + y * y + z * z;
        float* d = dvs + (lane * NA + a) * 4;
        d[0] = x; d[1] = y; d[2] = z; d[3] = r2;
    }

    // ---- Phase 1: evaluate all 512 primitives for this lane's sample ----
    // LDS layout bas[mt][p][row] is exactly the f32 WMMA A-operand feed.
    const int mt_l  = lane >> 4;
    const int row_l = lane & 15;
    const int pbase = mol * NP;
    for (int p = 0; p < NP; ++p) {
        int   c     = centers[pbase + p];          // uniform -> s_load
        int   valid = (c >= 0);
        int   ci    = valid ? c : 0;
        float alpha = exps[pbase + p];
        int   s     = syms[pbase + p];
        unsigned pk = SYMPK[s];
        const float* d = dvs + (lane * NA + ci) * 4;
        float x = d[0], y = d[1], z = d[2], r2 = d[3];
        float ang = powi3(x, pk & 0xFu) * powi3(y, (pk >> 4) & 0xFu)
                  * powi3(z, (pk >> 8) & 0xFu);
        float v = valid ? (ang * __expf(-alpha * r2)) : 0.0f;
        bas[(mt_l * NP + p) * 16 + row_l] = v;
    }
    __syncthreads();   // single-wave wg: barrier->NOP, but flushes DScnt

    // ---- Phase 2: T = B(16xK) * dm(KxN) via v_wmma_f32_16x16x4_f32,
    //      fused with rho[m] += T[m,q] * bas[m,q] via the C/D lane layout ----
    const float* dmN  = dm + (size_t)mol * NP * NP;
    const int koff  = (lane < 16) ? 0 : 2;   // A/B K-halves per 16-lane group
    const int nlane = lane & 15;             // N (and A-row M) index per lane
    const int mhalf = (lane < 16) ? 0 : 8;   // C/D: M = r + mhalf

    float part0[8], part1[8];
    #pragma unroll
    for (int r = 0; r < 8; ++r) { part0[r] = 0.0f; part1[r] = 0.0f; }

    for (int j = 0; j < NP / 16; ++j) {
        const int col = j * 16 + nlane;
        v8f acc0 = {}; v8f acc1 = {};
        const float* dmp = dmN + (size_t)koff * NP + col;  // += 4*NP per K-step

        #pragma unroll 4
        for (int k4 = 0; k4 < NP / 4; ++k4) {
            const int k0 = k4 * 4;
            // B operand: dm rows k0+koff, k0+koff+1 at column `col`
            v2f b;
            b.x = dmp[0];
            b.y = dmp[NP];
            dmp += 4 * NP;
            // A operands: bas rows (samples) nlane, K = k0+koff, +1
            const int ab = (k0 + koff) * 16 + nlane;
            v2f a0, a1;
            a0.x = bas[ab];
            a0.y = bas[ab + 16];
            a1.x = bas[NP * 16 + ab];
            a1.y = bas[NP * 16 + ab + 16];
            acc0 = __builtin_amdgcn_wmma_f32_16x16x4_f32(
                false, a0, false, b, (short)0, acc0, false, false);
            acc1 = __builtin_amdgcn_wmma_f32_16x16x4_f32(
                false, a1, false, b, (short)0, acc1, false, false);
        }

        // Stage 2: D layout lane l, reg r -> (M = r + mhalf, N = nlane).
        // part[m] += T[m, q=j*16+nlane] * bas[m, q]
        #pragma unroll
        for (int r = 0; r < 8; ++r) {
            const int m = r + mhalf;
            float b0 = bas[(j * 16 + nlane) * 16 + m];
            float b1 = bas[NP * 16 + (j * 16 + nlane) * 16 + m];
            part0[r] += acc0[r] * b0;
            part1[r] += acc1[r] * b1;
        }
    }

    // ---- Reduce partials within each 16-lane half and store rho ----
    float* outp = out + (size_t)mol * MS + sbase;
    #pragma unroll
    for (int r = 0; r < 8; ++r) {
        float v0 = part0[r];
        v0 += __shfl_xor(v0, 1, 32);
        v0 += __shfl_xor(v0, 2, 32);
        v0 += __shfl_xor(v0, 4, 32);
        v0 += __shfl_xor(v0, 8, 32);
        float v1 = part1[r];
        v1 += __shfl_xor(v1, 1, 32);
        v1 += __shfl_xor(v1, 2, 32);
        v1 += __shfl_xor(v1, 4, 32);
        v1 += __shfl_xor(v1, 8, 32);
        if (lane == 0) {
            outp[r]      = v0;   // M-tile 0, m = r
            outp[16 + r] = v1;   // M-tile 1, m = r
        }
        if (lane == 16) {
            outp[8 + r]      = v0;  // M-tile 0, m = 8 + r
            outp[16 + 8 + r] = v1;  // M-tile 1, m = 8 + r
        }
    }
}

extern "C" void kernel_launch(void* const* d_in, const int* in_sizes, int n_in,
                              void* d_out, int out_size, void* d_ws, size_t ws_size,
                              hipStream_t stream) {
    const float* dv      = (const float*)d_in[0];
    const int*   centers = (const int*)  d_in[1];
    const float* exps    = (const float*)d_in[2];
    const int*   syms    = (const int*)  d_in[3];
    const float* dm      = (const float*)d_in[4];
    float*       out     = (float*)d_out;

    const int grid = NMOL * TILEGRPS;                          // 2048 waves
    const size_t shmem = (size_t)(MT * NP * 16 + 32 * NA * 4) * sizeof(float); // 80 KB
    wfn_density_wmma_kernel<<<grid, 32, shmem, stream>>>(dv, centers, exps, syms,
                                                         dm, out);
}